// DomainAdaptationLoss_11931419148433
// MI455X (gfx1250) — compile-verified
//
#include <hip/hip_runtime.h>
#include <hip/hip_bf16.h>

typedef __attribute__((ext_vector_type(2))) float v2f;
typedef __attribute__((ext_vector_type(8))) float v8f;

#define SIGMA_C   (2.0f * 0.01f * 0.01f)   /* 2*sigma^2 = 2e-4 */
#define FSCORE_W  0.1f
#define TASK_W    1.0f
#define DOMAIN_W  0.1f

#define CHUNK 2048   /* target points staged per LDS pass: 2048*16B = 32 KB */

// ---------------------------------------------------------------------------
// Stage 1: fused Chamfer row-min via V_WMMA_F32_16X16X4_F32 with the entire
// distance computation folded into the matrix op:
//   A[m,:] = (-2x_p, -2y_p, -2z_p, 1)          (per 16-row query tile)
//   B[:,n] = ( x_t ,  y_t ,  z_t , |t_n|^2)    (staged in LDS as float4)
//   C[m,n] = |p_m|^2                           (broadcast along N lanes)
//   D = A.B + C = |p|^2 + |t|^2 - 2 p.t  == squared distance, directly.
// Inner loop per 16-column tile: 1x ds_load_b64, 1x wmma, 8x v_min.
//
// Fragment layouts (ISA 7.12.2, f32 16x16x4):
//   A: lanes 0-15 -> (K0,K1); lanes 16-31 -> (K2,K3).  B mirrored with N on
//   lanes -> lane reads float2 at LDS byte offset point*16 + (hi?8:0).
//   C/D: VGPR j, lanes 0-15 -> M=j, lanes 16-31 -> M=j+8; N = lane&15.
// ---------------------------------------------------------------------------
__global__ __launch_bounds__(256) void chamfer_min_wmma(
    const float* __restrict__ A,   // [B, NA, 3] query points
    const float* __restrict__ Bp,  // [B, NB, 3] database points
    float* __restrict__ out_min,   // [B, NA] min over database of sq dist
    int NA, int NB)
{
  __shared__ float tile[CHUNK * 4];   // packed (x, y, z, |t|^2)

  const int b    = blockIdx.y;
  const int wave = threadIdx.x >> 5;            // 0..7
  const int lane = threadIdx.x & 31;
  const int row0 = (blockIdx.x * 8 + wave) * 16;

  const float* __restrict__ Ab = A  + (size_t)b * NA * 3;
  const float* __restrict__ Bb = Bp + (size_t)b * NB * 3;

  const int  lrow = lane & 15;
  const bool hi   = lane >= 16;

  // A fragment with -2 folded in; K=3 slot = 1.0 multiplies |t|^2 from B.
  const float ax = Ab[(row0 + lrow) * 3 + 0];
  const float ay = Ab[(row0 + lrow) * 3 + 1];
  const float az = Ab[(row0 + lrow) * 3 + 2];
  v2f av;
  av[0] = hi ? (-2.f * az) : (-2.f * ax);
  av[1] = hi ? 1.0f       : (-2.f * ay);

  // C fragment: |p_M|^2 for the 8 rows this lane owns in D.
  const int rbase = row0 + (hi ? 8 : 0);
  v8f pm2;
#pragma unroll
  for (int j = 0; j < 8; ++j) {
    const float x = Ab[(rbase + j) * 3 + 0];
    const float y = Ab[(rbase + j) * 3 + 1];
    const float z = Ab[(rbase + j) * 3 + 2];
    pm2[j] = x * x + y * y + z * z;
  }

  float rowmin[8];
#pragma unroll
  for (int j = 0; j < 8; ++j) rowmin[j] = 3.4e38f;

  for (int base = 0; base < NB; base += CHUNK) {
    const int len = (NB - base < CHUNK) ? (NB - base) : CHUNK;

    // Cooperative stage: pack (x,y,z,|t|^2) into LDS, one float4 per point.
    for (int i = threadIdx.x; i < len; i += 256) {
      const float x = Bb[(size_t)(base + i) * 3 + 0];
      const float y = Bb[(size_t)(base + i) * 3 + 1];
      const float z = Bb[(size_t)(base + i) * 3 + 2];
      tile[i * 4 + 0] = x;
      tile[i * 4 + 1] = y;
      tile[i * 4 + 2] = z;
      tile[i * 4 + 3] = x * x + y * y + z * z;
    }
    __syncthreads();

    // This lane's B-fragment source: float2 at (x,y) or (z,|t|^2).
    const float* __restrict__ src = tile + (hi ? 2 : 0) + lrow * 4;
    const int ntiles = len >> 4;
    for (int ct = 0; ct < ntiles; ++ct) {
      const v2f bv = *(const v2f*)(src + ct * 64);   // (ct*16)*4 floats
      const v8f d = __builtin_amdgcn_wmma_f32_16x16x4_f32(
          /*neg_a=*/false, av, /*neg_b=*/false, bv,
          /*c_mod=*/(short)0, pm2, /*reuse_a=*/false, /*reuse_b=*/false);
#pragma unroll
      for (int j = 0; j < 8; ++j) rowmin[j] = fminf(rowmin[j], d[j]);
    }
    __syncthreads();
  }

  // Min across the 16 lanes of each half-wave.
#pragma unroll
  for (int m = 8; m >= 1; m >>= 1) {
#pragma unroll
    for (int j = 0; j < 8; ++j) {
      const float o = __shfl_xor(rowmin[j], m, 16);
      rowmin[j] = fminf(rowmin[j], o);
    }
  }

  if (lrow == 0) {
#pragma unroll
    for (int j = 0; j < 8; ++j)
      out_min[(size_t)b * NA + rbase + j] = rowmin[j];
  }
}

// ---------------------------------------------------------------------------
// Stage 2: per-batch sums of min and exp(-min/c) for both directions.
// sums[b*4 + {0,1,2,3}] = {sum(min_a), sum(soft_a), sum(min_b), sum(soft_b)}
// ---------------------------------------------------------------------------
__global__ __launch_bounds__(256) void reduce_batch(
    const float* __restrict__ min_a, const float* __restrict__ min_b,
    float* __restrict__ sums, int N, int M)
{
  const int b = blockIdx.x;
  const int t = threadIdx.x;
  float sa = 0.f, ea = 0.f, sb = 0.f, eb = 0.f;
  for (int i = t; i < N; i += 256) {
    const float v = min_a[(size_t)b * N + i];
    sa += v;
    ea += expf(-v / SIGMA_C);
  }
  for (int i = t; i < M; i += 256) {
    const float v = min_b[(size_t)b * M + i];
    sb += v;
    eb += expf(-v / SIGMA_C);
  }
  __shared__ float s0[256], s1[256], s2[256], s3[256];
  s0[t] = sa; s1[t] = ea; s2[t] = sb; s3[t] = eb;
  __syncthreads();
  for (int off = 128; off > 0; off >>= 1) {
    if (t < off) {
      s0[t] += s0[t + off];
      s1[t] += s1[t + off];
      s2[t] += s2[t + off];
      s3[t] += s3[t + off];
    }
    __syncthreads();
  }
  if (t == 0) {
    sums[b * 4 + 0] = s0[0];
    sums[b * 4 + 1] = s1[0];
    sums[b * 4 + 2] = s2[0];
    sums[b * 4 + 3] = s3[0];
  }
}

// ---------------------------------------------------------------------------
// Stage 3: tiny dense layer Y[s,j] = act(X[s,:] @ W[:,j] + bias[j])
// ---------------------------------------------------------------------------
__global__ __launch_bounds__(256) void dense_layer(
    const float* __restrict__ X, const float* __restrict__ W,
    const float* __restrict__ bias, float* __restrict__ Y,
    int S, int K, int Nout, int do_relu)
{
  const int idx = blockIdx.x * 256 + threadIdx.x;
  if (idx >= S * Nout) return;
  const int s = idx / Nout;
  const int j = idx - s * Nout;
  float acc = bias[j];
  for (int k = 0; k < K; ++k) acc += X[(size_t)s * K + k] * W[(size_t)k * Nout + j];
  if (do_relu) acc = fmaxf(acc, 0.f);
  Y[idx] = acc;
}

// ---------------------------------------------------------------------------
// Stage 4: combine everything into the scalar loss (tiny -> single thread).
// ---------------------------------------------------------------------------
__global__ void finalize(
    const float* __restrict__ sums,    // [B,4]
    const float* __restrict__ logits,  // [B,D]
    const int*   __restrict__ labels,  // [B]
    const float* __restrict__ dw,      // [B]
    float* __restrict__ out,
    int B, int N, int M, int D)
{
  if (threadIdx.x != 0 || blockIdx.x != 0) return;
  const float invN = 1.f / (float)N, invM = 1.f / (float)M, invB = 1.f / (float)B;

  float mma = 0.f, mmb = 0.f, mpa = 0.f, mpb = 0.f, dsw = 0.f;
  for (int b = 0; b < B; ++b) {
    const float ca = sums[b * 4 + 0] * invN;  // mean min_a (per sample)
    const float pa = sums[b * 4 + 1] * invN;  // p_i
    const float cb = sums[b * 4 + 2] * invM;
    const float pb = sums[b * 4 + 3] * invM;  // r_i
    mma += ca; mmb += cb; mpa += pa; mpb += pb;
    const float ch_i = ca + cb;
    const float f_i  = 2.f * pa * pb / (pa + pb + 1e-8f);
    const float loss_i = ch_i + FSCORE_W * (1.f - f_i);
    dsw += dw[b] * loss_i;
  }
  mma *= invB; mmb *= invB; mpa *= invB; mpb *= invB; dsw *= invB;

  const float chamfer = mma + mmb;
  const float fscore  = 2.f * mpa * mpb / (mpa + mpb + 1e-8f);
  const float task    = chamfer + FSCORE_W * (1.f - fscore);

  float domain = 0.f;
  for (int b = 0; b < B; ++b) {
    float mx = -3.4e38f;
    for (int d = 0; d < D; ++d) mx = fmaxf(mx, logits[b * D + d]);
    float se = 0.f;
    for (int d = 0; d < D; ++d) se += expf(logits[b * D + d] - mx);
    const float lse = mx + logf(se);
    const int lab = labels[b];
    domain -= (logits[b * D + lab] - lse);
  }
  domain *= invB;

  out[0] = TASK_W * task + DOMAIN_W * domain + dsw;
}

// ---------------------------------------------------------------------------
extern "C" void kernel_launch(void* const* d_in, const int* in_sizes, int n_in,
                              void* d_out, int out_size, void* d_ws, size_t ws_size,
                              hipStream_t stream) {
  const float* pred    = (const float*)d_in[0];   // [B,N,3]
  const float* target  = (const float*)d_in[1];   // [B,M,3]
  const float* dfeat   = (const float*)d_in[2];   // [B,F]
  const float* dw      = (const float*)d_in[3];   // [B]
  const float* W1      = (const float*)d_in[4];   // [F,H1]
  const float* b1      = (const float*)d_in[5];   // [H1]
  const float* W2      = (const float*)d_in[6];   // [H1,H2]
  const float* b2      = (const float*)d_in[7];   // [H2]
  const float* W3      = (const float*)d_in[8];   // [H2,D]
  const float* b3      = (const float*)d_in[9];   // [D]
  const int*   labels  = (const int*)d_in[10];    // [B]

  const int B  = in_sizes[3];
  const int N  = in_sizes[0] / (3 * B);
  const int M  = in_sizes[1] / (3 * B);
  const int F  = in_sizes[2] / B;
  const int H1 = in_sizes[5];
  const int H2 = in_sizes[7];
  const int D  = in_sizes[9];

  float* ws     = (float*)d_ws;
  float* min_a  = ws;                      // [B,N]
  float* min_b  = min_a + (size_t)B * N;   // [B,M]
  float* sums   = min_b + (size_t)B * M;   // [B,4]
  float* h1     = sums + B * 4;            // [B,H1]
  float* h2     = h1 + (size_t)B * H1;     // [B,H2]
  float* logits = h2 + (size_t)B * H2;     // [B,D]

  // Chamfer mins, both directions (128 rows per workgroup: 8 waves x 16 rows).
  {
    dim3 grid_a(N / 128, B), grid_b(M / 128, B);
    chamfer_min_wmma<<<grid_a, 256, 0, stream>>>(pred, target, min_a, N, M);
    chamfer_min_wmma<<<grid_b, 256, 0, stream>>>(target, pred, min_b, M, N);
  }

  // Per-batch sums.
  reduce_batch<<<B, 256, 0, stream>>>(min_a, min_b, sums, N, M);

  // Domain classifier MLP.
  {
    int n1 = B * H1, n2 = B * H2, n3 = B * D;
    dense_layer<<<(n1 + 255) / 256, 256, 0, stream>>>(dfeat, W1, b1, h1, B, F, H1, 1);
    dense_layer<<<(n2 + 255) / 256, 256, 0, stream>>>(h1, W2, b2, h2, B, H1, H2, 1);
    dense_layer<<<(n3 + 255) / 256, 256, 0, stream>>>(h2, W3, b3, logits, B, H2, D, 0);
  }

  // Scalar combine.
  finalize<<<1, 32, 0, stream>>>(sums, logits, labels, dw, (float*)d_out, B, N, M, D);
}